// SplineANN_46462956208149
// MI455X (gfx1250) — compile-verified
//
#include <hip/hip_runtime.h>
#include <cstdint>

// SplineANN for MI455X (gfx1250, wave32).
//
// Memory floor: x 16.8 MB + out 8.4 MB @ 23.3 TB/s ~= 1.1 us; the real limit
// is the coefficient gather: 131072*32*256 B = 1.07 GB of LDS reads.
// Design:
//   * table staged into LDS once per block via CDNA5 async global->LDS DMA
//     (ASYNCcnt path), then gathered with ds_load_b128.
//   * per-dim layout = 70 rows (67 + duplicates of rows 0..2) so the 4 spline
//     taps are physically consecutive -> no mod/wrap, all 16 ds loads per dim
//     use constant immediate offsets off one base address.
//   * rows padded to 20 floats (80 B): 16B-aligned b128 loads, row starts
//     spread over 16 LDS bank windows (vs 4 unpadded).
//   * v4 accumulators + __builtin_elementwise_fma -> v_pk_fma_f32 (VOP3P).
//   * all four spline weights evaluated with ONE packed Horner recurrence
//     (3 vector FMAs) instead of ~12 scalar ops.
// WMMA is deliberately not used: the gathered B-matrix is batch-dependent;
// the dense [B x 2144][2144 x 16] WMMA reformulation wastes 16x MACs and
// needs per-fragment sparse-A construction -> strictly slower than this.

#define BATCH      131072
#define IN_DIM     32
#define SUBS       64
#define DENSITY    67
#define DROWS      70                        // 67 + 3 duplicated wrap rows
#define OUT_DIM    16
#define NLROWS     (IN_DIM * DROWS)          // 2240 logical LDS rows
#define LDS_STRIDE 20                        // floats/row: 80 B
#define LDS_BYTES  (NLROWS * LDS_STRIDE * 4) // 179,200 B
#define BLOCK      512
#define GRID       (BATCH / BLOCK)           // 256 -> one lane per batch row

typedef float v4f __attribute__((ext_vector_type(4)));

__global__ __launch_bounds__(BLOCK, 1)
void spline_ann_kernel(const float* __restrict__ x,
                       const float* __restrict__ table,
                       float* __restrict__ out)
{
    extern __shared__ float tab[];           // NLROWS * LDS_STRIDE floats

    // ---- Stage table into LDS via CDNA5 async global->LDS DMA.
    // Logical row lrow = d*70 + i maps to source row d*67 + (i mod 67):
    // rows 67..69 of each dim duplicate rows 0..2 (handles the tap wrap,
    // since base = floor(x*64) <= 64 -> base+3 <= 67 <= 69).
    for (int lrow = threadIdx.x; lrow < NLROWS; lrow += BLOCK) {
        int d = lrow / DROWS;
        int i = lrow - d * DROWS;
        int srow = d * DENSITY + (i >= DENSITY ? i - DENSITY : i);
        const float* src = table + srow * OUT_DIM;            // 64 B per row
        uint32_t dst = (uint32_t)(uintptr_t)(&tab[lrow * LDS_STRIDE]);
        // INST_OFFSET is added to both the LDS and global addresses.
        asm volatile(
            "global_load_async_to_lds_b128 %0, %1, off\n\t"
            "global_load_async_to_lds_b128 %0, %1, off offset:16\n\t"
            "global_load_async_to_lds_b128 %0, %1, off offset:32\n\t"
            "global_load_async_to_lds_b128 %0, %1, off offset:48"
            :: "v"(dst), "v"(src) : "memory");
    }
    asm volatile("s_wait_asynccnt 0" ::: "memory");
    __syncthreads();

    // ---- One batch row per lane; coalesced 128B x-read, 64B out-write ----
    const int r = blockIdx.x * BLOCK + threadIdx.x;          // 0..131071
    const v4f* __restrict__ x4 =
        reinterpret_cast<const v4f*>(x + (size_t)r * IN_DIM);

    v4f acc0 = {0.0f, 0.0f, 0.0f, 0.0f};
    v4f acc1 = {0.0f, 0.0f, 0.0f, 0.0f};
    v4f acc2 = {0.0f, 0.0f, 0.0f, 0.0f};
    v4f acc3 = {0.0f, 0.0f, 0.0f, 0.0f};

    // Packed Horner coefficients for the 4 B-spline basis weights,
    // tap order {p4, p3, p2, p1} pairing rows base+0 .. base+3.
    const v4f A3 = {-1.0f / 6.0f,  0.5f, -0.5f, 1.0f / 6.0f};
    const v4f A2 = { 0.5f,        -1.0f,  0.5f, 0.0f       };
    const v4f A1 = {-0.5f,         0.0f,  0.5f, 0.0f       };
    const v4f A0 = { 1.0f / 6.0f,  2.0f / 3.0f, 1.0f / 6.0f, 0.0f};

    for (int dq = 0; dq < IN_DIM / 4; ++dq) {
        v4f xq = x4[dq];                                     // global_load_b128
        #pragma unroll
        for (int j = 0; j < 4; ++j) {
            const int d = dq * 4 + j;
            float xs = xq[j] * (float)SUBS;                  // [0, 64]
            float fl = floorf(xs);
            float t  = xs - fl;
            v4f tv = {t, t, t, t};
            // w = ((A3*t + A2)*t + A1)*t + A0  -> 3 packed FMAs for all taps
            v4f w4 = __builtin_elementwise_fma(A3, tv, A2);
            w4     = __builtin_elementwise_fma(w4, tv, A1);
            w4     = __builtin_elementwise_fma(w4, tv, A0);
            int base = (int)fl;                              // 0..64, no wrap
            // 4 consecutive padded rows; 16 ds_load_b128 with imm offsets.
            const v4f* p = reinterpret_cast<const v4f*>(
                tab + (d * DROWS + base) * LDS_STRIDE);
            #pragma unroll
            for (int k = 0; k < 4; ++k) {
                v4f wv = {w4[k], w4[k], w4[k], w4[k]};
                v4f c0 = p[k * (LDS_STRIDE / 4) + 0];        // offset k*80+0
                v4f c1 = p[k * (LDS_STRIDE / 4) + 1];        // offset k*80+16
                v4f c2 = p[k * (LDS_STRIDE / 4) + 2];        // offset k*80+32
                v4f c3 = p[k * (LDS_STRIDE / 4) + 3];        // offset k*80+48
                acc0 = __builtin_elementwise_fma(wv, c0, acc0);  // v_pk_fma_f32
                acc1 = __builtin_elementwise_fma(wv, c1, acc1);
                acc2 = __builtin_elementwise_fma(wv, c2, acc2);
                acc3 = __builtin_elementwise_fma(wv, c3, acc3);
            }
        }
    }

    v4f* o4 = reinterpret_cast<v4f*>(out + (size_t)r * OUT_DIM);
    o4[0] = acc0;
    o4[1] = acc1;
    o4[2] = acc2;
    o4[3] = acc3;
}

extern "C" void kernel_launch(void* const* d_in, const int* in_sizes, int n_in,
                              void* d_out, int out_size, void* d_ws, size_t ws_size,
                              hipStream_t stream) {
    (void)in_sizes; (void)n_in; (void)out_size; (void)d_ws; (void)ws_size;
    const float* x     = (const float*)d_in[0];   // [131072, 32] f32
    const float* table = (const float*)d_in[1];   // [2144, 16]  f32
    float* out = (float*)d_out;                   // [131072, 16] f32
    spline_ann_kernel<<<dim3(GRID), dim3(BLOCK), LDS_BYTES, stream>>>(x, table, out);
}